// KnowledgeAwareGraphNetworks_29738353558006
// MI455X (gfx1250) — compile-verified
//
#include <hip/hip_runtime.h>
#include <hip/hip_bf16.h>
#include <math.h>

// D = 128 feature dim throughout.
#define FD 128
#define WS_STRIDE 132   // LDS row stride (floats): 132 % 64 = 4 -> conflict-free f32x2 reads

typedef __attribute__((ext_vector_type(2))) float v2f;
typedef __attribute__((ext_vector_type(8))) float v8f;

// --- h0[n][:] = emb[ids[n]][:] : one wave per node, float4 per lane ---
__global__ __launch_bounds__(256) void kag_gather(const int* __restrict__ ids,
                                                  const float* __restrict__ emb,
                                                  float* __restrict__ h, int n_nodes) {
    int lane = threadIdx.x & 31;
    int node = blockIdx.x * 8 + (threadIdx.x >> 5);
    if (node >= n_nodes) return;
    int c = ids[node];
    const float4* s = (const float4*)(emb + (size_t)c * FD);
    float4* d = (float4*)(h + (size_t)node * FD);
    d[lane] = s[lane];
}

// --- agg[dst[e]] += h[src[e]] : one wave per edge, float4 load + 4 f32 atomics per lane ---
__global__ __launch_bounds__(256) void kag_scatter(const float* __restrict__ h,
                                                   const int* __restrict__ src,
                                                   const int* __restrict__ dst,
                                                   float* __restrict__ agg, int n_edges) {
    int lane = threadIdx.x & 31;
    int e = blockIdx.x * 8 + (threadIdx.x >> 5);
    if (e >= n_edges) return;
    int s = src[e];
    int d = dst[e];
    float4 v = ((const float4*)(h + (size_t)s * FD))[lane];
    float* a = agg + (size_t)d * FD + lane * 4;
    atomicAdd(a + 0, v.x);
    atomicAdd(a + 1, v.y);
    atomicAdd(a + 2, v.z);
    atomicAdd(a + 3, v.w);
}

// --- out = relu(A @ W^T + bias), A:[n,128], W:[128,128] row-major ---
// One wave computes a 16x128 output slab via V_WMMA_F32_16X16X4_F32 (K-loop of 32).
__global__ __launch_bounds__(256) void kag_gemm_relu(const float* __restrict__ A,
                                                     const float* __restrict__ W,
                                                     const float* __restrict__ bias,
                                                     float* __restrict__ out, int n_nodes) {
    __shared__ float Ws[FD * WS_STRIDE];  // W[j][k] at Ws[j*WS_STRIDE + k], padded vs banks
    __shared__ float bs[FD];

    for (int i = threadIdx.x; i < FD * FD / 4; i += 256) {
        int row = i / (FD / 4);
        int c4  = i % (FD / 4);
        float4 v = ((const float4*)W)[i];
        float* p = &Ws[row * WS_STRIDE + c4 * 4];
        p[0] = v.x; p[1] = v.y; p[2] = v.z; p[3] = v.w;
    }
    if (threadIdx.x < FD) bs[threadIdx.x] = bias[threadIdx.x];
    __syncthreads();

    int lane = threadIdx.x & 31;
    int tile = blockIdx.x * 8 + (threadIdx.x >> 5);   // 16-row M tile
    if (tile * 16 >= n_nodes) return;                 // whole-wave guard: EXEC all-ones for WMMA
    int m0    = tile * 16;
    int mrow  = lane & 15;          // A: row within tile / B: column within tile
    int khalf = (lane >> 4) << 1;   // 0 or 2: which K pair this half-wave holds

    const float* Arow = A + (size_t)(m0 + mrow) * FD + khalf;

    v8f acc[8] = {};                // 8 N-tiles of 16 cols -> full 128-wide output row block
    for (int kb = 0; kb < 32; ++kb) {
        int k0 = kb * 4;
        v2f a;
        a.x = Arow[k0];
        a.y = Arow[k0 + 1];
#pragma unroll
        for (int nt = 0; nt < 8; ++nt) {
            const float* bp = &Ws[(nt * 16 + mrow) * WS_STRIDE + k0 + khalf];  // B = W^T
            v2f b;
            b.x = bp[0];
            b.y = bp[1];
            acc[nt] = __builtin_amdgcn_wmma_f32_16x16x4_f32(
                false, a, false, b, (short)0, acc[nt], false, false);
        }
    }

    // D layout: VGPR r -> row (m0 + r + 8*(lane>=16)), col = nt*16 + lane%16
    int rofs = (lane >> 4) << 3;
#pragma unroll
    for (int nt = 0; nt < 8; ++nt) {
        int col = nt * 16 + mrow;
        float bv = bs[col];
#pragma unroll
        for (int r = 0; r < 8; ++r) {
            float v = acc[nt][r] + bv;
            out[(size_t)(m0 + rofs + r) * FD + col] = fmaxf(v, 0.0f);
        }
    }
}

// --- out[n] = sigmoid(dot(h[n], Wout) + bout) : one wave per node, shuffle reduce ---
__global__ __launch_bounds__(256) void kag_head(const float* __restrict__ h,
                                                const float* __restrict__ Wout,
                                                const float* __restrict__ bout,
                                                float* __restrict__ out, int n_nodes) {
    __shared__ float w[FD];
    if (threadIdx.x < FD) w[threadIdx.x] = Wout[threadIdx.x];
    __syncthreads();
    int lane = threadIdx.x & 31;
    int node = blockIdx.x * 8 + (threadIdx.x >> 5);
    if (node >= n_nodes) return;
    float4 v = ((const float4*)(h + (size_t)node * FD))[lane];
    const float* wp = &w[lane * 4];
    float s = v.x * wp[0] + v.y * wp[1] + v.z * wp[2] + v.w * wp[3];
#pragma unroll
    for (int off = 16; off > 0; off >>= 1) s += __shfl_down(s, off, 32);
    if (lane == 0) out[node] = 1.0f / (1.0f + __expf(-(s + bout[0])));
}

extern "C" void kernel_launch(void* const* d_in, const int* in_sizes, int n_in,
                              void* d_out, int out_size, void* d_ws, size_t ws_size,
                              hipStream_t stream) {
    const int*   cncpt = (const int*)d_in[0];
    const int*   src   = (const int*)d_in[1];
    const int*   dst   = (const int*)d_in[2];
    const float* emb   = (const float*)d_in[3];
    const float* W1    = (const float*)d_in[4];
    const float* b1    = (const float*)d_in[5];
    const float* W2    = (const float*)d_in[6];
    const float* b2    = (const float*)d_in[7];
    const float* Wout  = (const float*)d_in[8];
    const float* bout  = (const float*)d_in[9];
    float* out = (float*)d_out;

    int n_nodes = in_sizes[0];
    int n_edges = in_sizes[1];

    size_t feat_elems = (size_t)n_nodes * FD;
    float* bufA = (float*)d_ws;          // node features (51.2 MB)
    float* bufB = bufA + feat_elems;     // aggregation buffer (51.2 MB)

    int node_blocks = (n_nodes + 7) / 8;
    int edge_blocks = (n_edges + 7) / 8;
    int tiles       = (n_nodes + 15) / 16;
    int gemm_blocks = (tiles + 7) / 8;

    // h0 = emb[ids]
    kag_gather<<<node_blocks, 256, 0, stream>>>(cncpt, emb, bufA, n_nodes);

    // layer 1
    hipMemsetAsync(bufB, 0, feat_elems * sizeof(float), stream);
    kag_scatter<<<edge_blocks, 256, 0, stream>>>(bufA, src, dst, bufB, n_edges);
    kag_gemm_relu<<<gemm_blocks, 256, 0, stream>>>(bufB, W1, b1, bufA, n_nodes);

    // layer 2
    hipMemsetAsync(bufB, 0, feat_elems * sizeof(float), stream);
    kag_scatter<<<edge_blocks, 256, 0, stream>>>(bufA, src, dst, bufB, n_edges);
    kag_gemm_relu<<<gemm_blocks, 256, 0, stream>>>(bufB, W2, b2, bufA, n_nodes);

    // sigmoid head
    kag_head<<<node_blocks, 256, 0, stream>>>(bufA, Wout, bout, out, n_nodes);
}